// Expert_WNO_68719476863
// MI455X (gfx1250) — compile-verified
//
#include <hip/hip_runtime.h>
#include <cstddef>

// ---------------------------------------------------------------------------
// Expert WNO (wavelet neural operator) for MI455X / gfx1250.
// BATCH=16, WIDTH=64, SIZE=8192, LEVEL=4, NEXP=10 (db1..db10), MMAX=529.
// Channel-mix einsum done with V_WMMA_F32_16X16X4_F32 (full fp32 precision;
// problem is HBM-bound so low-precision WMMA would not help).
// ---------------------------------------------------------------------------

#define NEXP   10
#define MMAX   529
#define ROWS   1024      // BATCH * WIDTH
#define SIZE0  8192

typedef float v2f __attribute__((ext_vector_type(2)));
typedef float v8f __attribute__((ext_vector_type(8)));

// pywt rec_lo (reconstruction low-pass) for db1..db10, natural order h[0..L-1].
// dec filters and rec_hi are derived: g[k] = (-1)^k h[L-1-k].
__constant__ float REC_LO[NEXP][20] = {
  {0.7071067811865476f, 0.7071067811865476f},
  {0.48296291314469025f, 0.8365163037378079f, 0.2241438680420134f, -0.12940952255126037f},
  {0.3326705529509569f, 0.8068915093133388f, 0.4598775021193313f, -0.13501102001039084f,
   -0.08544127388224149f, 0.035226291882100656f},
  {0.23037781330885523f, 0.7148465705525415f, 0.6308807679295904f, -0.02798376941698385f,
   -0.18703481171888114f, 0.030841381835986965f, 0.032883011666982945f, -0.010597401784997278f},
  {0.160102397974125f, 0.6038292697974729f, 0.7243085284385744f, 0.13842814590110342f,
   -0.24229488706619015f, -0.03224486958502952f, 0.07757149384006515f, -0.006241490213011705f,
   -0.012580751999015526f, 0.003335725285001549f},
  {0.11154074335008017f, 0.4946238903983854f, 0.7511339080215775f, 0.3152503517092432f,
   -0.22626469396516913f, -0.12976686756709563f, 0.09750160558707936f, 0.02752286553001629f,
   -0.031582039318031156f, 0.0005538422009938016f, 0.004777257511010651f, -0.00107730108499558f},
  {0.07785205408506236f, 0.39653931948230575f, 0.7291320908465551f, 0.4697822874053586f,
   -0.14390600392910627f, -0.22403618499416572f, 0.07130921926705004f, 0.08061260915107307f,
   -0.03802993693503463f, -0.01657454163101562f, 0.012550998556013784f, 0.00042957797300470274f,
   -0.0018016407039998328f, 0.0003537138000010399f},
  {0.05441584224308161f, 0.3128715909144659f, 0.6756307362980128f, 0.5853546836548691f,
   -0.015829105256023893f, -0.2840155429624281f, 0.00047248457399797254f, 0.128747426620186f,
   -0.01736930100202211f, -0.04408825393106472f, 0.013981027917015516f, 0.008746094047015655f,
   -0.00487035299301066f, -0.0003917403729959771f, 0.0006754494059985568f, -0.00011747678400228192f},
  {0.03807794736316728f, 0.24383467463766728f, 0.6048231236767786f, 0.6572880780366389f,
   0.13319738582208895f, -0.29327378327258685f, -0.09684078322087904f, 0.14854074933476008f,
   0.030725681478322865f, -0.06763282905952399f, 0.00025094711499193845f, 0.022361662123515244f,
   -0.004723204757894831f, -0.004281503681904723f, 0.0018476468829611268f, 0.00023038576399541288f,
   -0.0002519631889981789f, 3.9347319995026124e-05f},
  {0.026670057900950818f, 0.18817680007762133f, 0.5272011889309198f, 0.6884590394525921f,
   0.2811723436604265f, -0.24984642432648865f, -0.19594627437659665f, 0.12736934033574265f,
   0.09305736460380659f, -0.07139414716586077f, -0.02945753682194567f, 0.03321267405893324f,
   0.0036065535669883944f, -0.010733175482979604f, 0.0013953517469940798f, 0.00199240529499085f,
   -0.0006858566950046825f, -0.0001164668549943862f, 9.358867000108985e-05f, -1.326420300235487e-05f},
};

__device__ __forceinline__ int reflect_idx(int m, int N) {
  if (m < 0)       m = -m - 1;           // half-sample symmetric (pads < N)
  else if (m >= N) m = 2 * N - 1 - m;
  return m;
}

// One DWT level: a[j] = sum_t xp[2j+t]*h[t], d[j] = sum_t xp[2j+t]*g[t],
// xp = symmetric pad of x with L-2 on the left.
__global__ __launch_bounds__(256)
void dwt_kernel(const float* __restrict__ X, int N, int strideX,
                float* __restrict__ Ao, float* __restrict__ Do,
                int n_out, int strideOut, int e, int L) {
  const int j   = blockIdx.x * blockDim.x + threadIdx.x;
  const int row = blockIdx.y;
  if (j >= n_out) return;
  const float* x = X + (size_t)row * strideX;
  float sa = 0.f, sd = 0.f;
  const int base = 2 * j - (L - 2);
  for (int t = 0; t < L; ++t) {
    const float xv = x[reflect_idx(base + t, N)];
    const float h  = REC_LO[e][t];
    const float hr = REC_LO[e][L - 1 - t];
    const float g  = (t & 1) ? -hr : hr;
    sa += xv * h;
    sd += xv * g;
  }
  Ao[(size_t)row * strideOut + j] = sa;
  Do[(size_t)row * strideOut + j] = sd;
}

// Channel mixing at coarsest level: O[b,o,x] = sum_i A[b,i,x] * W[i,o,x].
// Per x: (16x64)*(64x64) matmul -> 4 N-tiles x 16 K-steps of wmma f32 16x16x4.
// One wave per (x, o-tile). EXEC is all-ones inside the wave (uniform guard).
__global__ __launch_bounds__(256)
void mix_wmma_kernel(const float* __restrict__ A, const float* __restrict__ W,
                     float* __restrict__ O, int n4, int strideA, int strideO) {
  const int waveInBlock = threadIdx.x >> 5;
  const int lane = threadIdx.x & 31;
  const int unit = blockIdx.x * 8 + waveInBlock;   // 8 waves / 256-thread block
  if (unit >= n4 * 4) return;                      // wave-uniform exit
  const int x    = unit >> 2;
  const int ot   = unit & 3;                       // 16-wide output-channel tile
  const int half = lane >> 4;                      // 0: K even-pair 0, 1: K pair 2
  const int l16  = lane & 15;
  const int m    = l16;                            // A row (batch)
  const int n    = ot * 16 + l16;                  // output channel (B/C column)

  v8f acc = {};
#pragma unroll
  for (int ks = 0; ks < 16; ++ks) {
    const int kb = ks * 4 + half * 2;              // this lane holds K = kb, kb+1
    v2f a, b;
    a[0] = A[((size_t)(m * 64 + kb)     ) * strideA + x];
    a[1] = A[((size_t)(m * 64 + kb + 1) ) * strideA + x];
    b[0] = W[((size_t)(kb)     * 64 + n) * MMAX + x];
    b[1] = W[((size_t)(kb + 1) * 64 + n) * MMAX + x];
    acc = __builtin_amdgcn_wmma_f32_16x16x4_f32(
        /*neg_a=*/false, a, /*neg_b=*/false, b,
        /*c_mod=*/(short)0, acc, /*reuse_a=*/false, /*reuse_b=*/false);
  }
#pragma unroll
  for (int r = 0; r < 8; ++r) {                    // C/D layout: vgpr r -> M = half*8 + r
    const int brow = half * 8 + r;
    O[((size_t)(brow * 64 + n)) * strideO + x] = acc[r];
  }
}

// One inverse DWT level: r[s] = sum_j a[j]*h[m] + d[j]*g[m], m = s+L-2-2j in [0,L-1].
__global__ __launch_bounds__(256)
void idwt_kernel(const float* __restrict__ A, int strideA,
                 const float* __restrict__ D, int strideD,
                 float* __restrict__ R, int strideR,
                 int nd, int rec_len, int e, int L) {
  const int s   = blockIdx.x * blockDim.x + threadIdx.x;
  const int row = blockIdx.y;
  if (s >= rec_len) return;
  const float* a = A + (size_t)row * strideA;
  const float* d = D + (size_t)row * strideD;
  const int jmin = s >> 1;                               // ceil((s-1)/2)
  int jmax = (s + L - 2) >> 1;
  if (jmax > nd - 1) jmax = nd - 1;
  float sum = 0.f;
  for (int j = jmin; j <= jmax; ++j) {
    const int mm = s + L - 2 - 2 * j;
    const float h  = REC_LO[e][mm];
    const float hr = REC_LO[e][L - 1 - mm];
    const float g  = (mm & 1) ? -hr : hr;
    sum += a[j] * h + d[j] * g;
  }
  R[(size_t)row * strideR + s] = sum;
}

// Final level fused with lambda-scaled accumulation into the output.
// (final rec_len == 8192 exactly for every even L)
__global__ __launch_bounds__(256)
void idwt_final_kernel(const float* __restrict__ A, int strideA,
                       const float* __restrict__ D, int strideD,
                       const float* __restrict__ lam, float* __restrict__ out,
                       int nd, int e, int L, int firstExpert) {
  const int s   = blockIdx.x * blockDim.x + threadIdx.x;
  const int row = blockIdx.y;
  if (s >= SIZE0) return;
  const float* a = A + (size_t)row * strideA;
  const float* d = D + (size_t)row * strideD;
  const int jmin = s >> 1;
  int jmax = (s + L - 2) >> 1;
  if (jmax > nd - 1) jmax = nd - 1;
  float sum = 0.f;
  for (int j = jmin; j <= jmax; ++j) {
    const int mm = s + L - 2 - 2 * j;
    const float h  = REC_LO[e][mm];
    const float hr = REC_LO[e][L - 1 - mm];
    const float g  = (mm & 1) ? -hr : hr;
    sum += a[j] * h + d[j] * g;
  }
  const float lv = lam[(size_t)row * NEXP + e];
  const size_t o = (size_t)row * SIZE0 + s;
  if (firstExpert) out[o] = lv * sum;
  else             out[o] += lv * sum;
}

static inline int cdiv(int a, int b) { return (a + b - 1) / b; }

extern "C" void kernel_launch(void* const* d_in, const int* in_sizes, int n_in,
                              void* d_out, int out_size, void* d_ws, size_t ws_size,
                              hipStream_t stream) {
  (void)in_sizes; (void)n_in; (void)out_size; (void)ws_size;
  const float* x   = (const float*)d_in[0];   // (16, 64, 8192)
  const float* lam = (const float*)d_in[1];   // (16, 64, 10)
  const float* w1  = (const float*)d_in[2];   // (10, 64, 64, 529)
  const float* w2  = (const float*)d_in[3];   // (10, 64, 64, 529)
  float* out = (float*)d_out;                 // (16, 64, 8192)

  // Workspace layout (row strides in floats); recon buffers alias dead approx buffers.
  const int W1L = 4128, W2L = 2080, W3L = 1056, W4L = 544;
  float* p = (float*)d_ws;
  float* A1  = p; p += (size_t)ROWS * W1L;    // level-1 approx, later R1
  float* D1  = p; p += (size_t)ROWS * W1L;
  float* A2  = p; p += (size_t)ROWS * W2L;    // level-2 approx, later R2
  float* D2  = p; p += (size_t)ROWS * W2L;
  float* A3  = p; p += (size_t)ROWS * W3L;    // level-3 approx, later R3
  float* D3  = p; p += (size_t)ROWS * W3L;
  float* A4  = p; p += (size_t)ROWS * W4L;
  float* D4  = p; p += (size_t)ROWS * W4L;
  float* A4M = p; p += (size_t)ROWS * W4L;    // channel-mixed coarse approx
  float* D4M = p; p += (size_t)ROWS * W4L;    // channel-mixed coarse detail

  const dim3 blk(256);
  for (int e = 0; e < NEXP; ++e) {
    const int L  = 2 * (e + 1);
    const int n1 = (SIZE0 + L - 1) / 2;
    const int n2 = (n1 + L - 1) / 2;
    const int n3 = (n2 + L - 1) / 2;
    const int n4 = (n3 + L - 1) / 2;

    // ---- analysis: 4 DWT levels ----
    dwt_kernel<<<dim3(cdiv(n1, 256), ROWS), blk, 0, stream>>>(x,  SIZE0, SIZE0, A1, D1, n1, W1L, e, L);
    dwt_kernel<<<dim3(cdiv(n2, 256), ROWS), blk, 0, stream>>>(A1, n1, W1L, A2, D2, n2, W2L, e, L);
    dwt_kernel<<<dim3(cdiv(n3, 256), ROWS), blk, 0, stream>>>(A2, n2, W2L, A3, D3, n3, W3L, e, L);
    dwt_kernel<<<dim3(cdiv(n4, 256), ROWS), blk, 0, stream>>>(A3, n3, W3L, A4, D4, n4, W4L, e, L);

    // ---- channel mixing via f32 WMMA (per x: 16x64 @ 64x64) ----
    const int units = 4 * n4;                 // 4 o-tiles per position
    mix_wmma_kernel<<<cdiv(units, 8), 256, 0, stream>>>(A4, w1 + (size_t)e * 64 * 64 * MMAX, A4M, n4, W4L, W4L);
    mix_wmma_kernel<<<cdiv(units, 8), 256, 0, stream>>>(D4, w2 + (size_t)e * 64 * 64 * MMAX, D4M, n4, W4L, W4L);

    // ---- synthesis: 4 IDWT levels (last fused with lambda-accumulate) ----
    const int r3len = 2 * n4 - L + 2;
    idwt_kernel<<<dim3(cdiv(r3len, 256), ROWS), blk, 0, stream>>>(A4M, W4L, D4M, W4L, A3, W3L, n4, r3len, e, L);
    const int r2len = 2 * n3 - L + 2;
    idwt_kernel<<<dim3(cdiv(r2len, 256), ROWS), blk, 0, stream>>>(A3, W3L, D3, W3L, A2, W2L, n3, r2len, e, L);
    const int r1len = 2 * n2 - L + 2;
    idwt_kernel<<<dim3(cdiv(r1len, 256), ROWS), blk, 0, stream>>>(A2, W2L, D2, W2L, A1, W1L, n2, r1len, e, L);
    idwt_final_kernel<<<dim3(cdiv(SIZE0, 256), ROWS), blk, 0, stream>>>(A1, W1L, D1, W1L, lam, out, n1, e, L, e == 0);
  }
}